// Log2Quantizer_61718680043579
// MI455X (gfx1250) — compile-verified
//
#include <hip/hip_runtime.h>

// Log2Quantizer for x: (8, 12, 4096, 64) fp32.
// Live math only (the reference's sort/std/rank pipeline is dead code:
// bit_token is overwritten with N_BITS, so n_levels == 256 everywhere):
//   delta[b,t] = max over (h,c) of x[b,:,t,:]
//   ratio      = max(x/delta, 1e-8)
//   e          = round_half_even(-log2(ratio))
//   out        = (e >= 256) ? 0 : delta * 2^(-clip(e, 0, 255))
//
// Memory-bound: 100 MB in + 100 MB out -> ~8.6 us at 23.3 TB/s.
// Single fused pass: gfx1250 async global->LDS staging (ASYNCcnt-pipelined,
// no workgroup barriers needed since each thread consumes only its own
// staged column), wave32 shuffle butterfly for the per-(b,t) max, then a
// ~10-VALU/element transform (v_mul, v_max, v_log_f32, v_rndne, clamps,
// v_cvt, v_ldexp, v_cndmask).

namespace {
constexpr int Bb = 8;
constexpr int Hh = 12;
constexpr int Tt = 4096;
constexpr int Cc = 64;
constexpr int TT = 16;                 // t-rows per block tile
constexpr int NTHREADS = 256;          // 8 waves (wave32)
constexpr int TILE_F = Hh * TT * Cc;   // 12288 floats = 48 KB LDS staging
}

__device__ __forceinline__ float quant_one(float v, float inv_d, float d) {
  float r  = fmaxf(v * inv_d, 1e-8f);
  float en = rintf(__log2f(r));                 // en = -x_int, in [-27, 0];
                                                // rint(-x) == -rint(x) under RNE
  float qn = fminf(fmaxf(en, -255.0f), 0.0f);   // qn = -clip(x_int, 0, 255)
  float res = ldexpf(d, (int)qn);               // exact delta * 2^-q
  return (en <= -256.0f) ? 0.0f : res;          // mask = x_int >= n_levels
}

__global__ __launch_bounds__(NTHREADS)
void log2quant_kernel(const float* __restrict__ x, float* __restrict__ out) {
  __shared__ float tile[TILE_F];   // [h][t_local][c] staging only (no sharing)

  const int tid = threadIdx.x;
  const int b   = blockIdx.x / (Tt / TT);
  const int t0  = (blockIdx.x % (Tt / TT)) * TT;
  const int c4  = tid & 15;   // float4 column within the 64-float c row
  const int tl  = tid >> 4;   // t-row within tile, 0..15

  // This thread's element-0 position in the (b,h,t,c) tensor, h == 0.
  const size_t gelem0 =
      ((size_t)(b * Hh) * Tt + (size_t)(t0 + tl)) * Cc + (size_t)c4 * 4;
  const char* gbase = reinterpret_cast<const char*>(x) + gelem0 * sizeof(float);
  const int lbase_f = tl * Cc + c4 * 4;
  // addrspace(3) offset == low 32 bits of the flat LDS pointer
  const unsigned lbyte = (unsigned)(size_t)&tile[lbase_f];

  // ---- Issue 12 async global->LDS 16B copies (one per h) ----
#pragma unroll
  for (int h = 0; h < Hh; ++h) {
    const char* gp = gbase + (size_t)h * (size_t)Tt * Cc * sizeof(float);
    unsigned    lp = lbyte + h * (TT * Cc * (int)sizeof(float));
    asm volatile("global_load_async_to_lds_b128 %0, %1, off"
                 :: "v"(lp), "v"(gp)
                 : "memory");
  }

  // ---- Consume chunks as they land ----
  // Async loads return "done" in order, so ASYNCcnt <= (11-h) guarantees
  // chunks 0..h have been written to LDS. "memory" clobber keeps the
  // ds_load from being scheduled above the wait.
  float4 vals[Hh];
  float  m = 0.0f;   // x in [0,1): 0 is a safe identity for max

#define CONSUME(h)                                                             \
  asm volatile("s_wait_asynccnt %0" ::"n"(Hh - 1 - (h)) : "memory");           \
  vals[h] = *reinterpret_cast<const float4*>(&tile[lbase_f + (h) * (TT * Cc)]);\
  m = fmaxf(m, fmaxf(fmaxf(vals[h].x, vals[h].y),                              \
                     fmaxf(vals[h].z, vals[h].w)));
  CONSUME(0)  CONSUME(1)  CONSUME(2)  CONSUME(3)
  CONSUME(4)  CONSUME(5)  CONSUME(6)  CONSUME(7)
  CONSUME(8)  CONSUME(9)  CONSUME(10) CONSUME(11)
#undef CONSUME

  // ---- 16-lane wave32 butterfly: every lane ends with the row max ----
  m = fmaxf(m, __shfl_xor(m, 8, 32));
  m = fmaxf(m, __shfl_xor(m, 4, 32));
  m = fmaxf(m, __shfl_xor(m, 2, 32));
  m = fmaxf(m, __shfl_xor(m, 1, 32));

  const float d     = m;
  const float inv_d = 1.0f / d;   // one precise divide amortized over 48 elems

  // ---- Transform register-resident data, coalesced float4 stores ----
  float4* out4 = reinterpret_cast<float4*>(out);
  const size_t gidx4 = gelem0 / 4;
#pragma unroll
  for (int h = 0; h < Hh; ++h) {
    float4 o;
    o.x = quant_one(vals[h].x, inv_d, d);
    o.y = quant_one(vals[h].y, inv_d, d);
    o.z = quant_one(vals[h].z, inv_d, d);
    o.w = quant_one(vals[h].w, inv_d, d);
    out4[gidx4 + (size_t)h * (size_t)(Tt * Cc / 4)] = o;
  }
}

extern "C" void kernel_launch(void* const* d_in, const int* in_sizes, int n_in,
                              void* d_out, int out_size, void* d_ws, size_t ws_size,
                              hipStream_t stream) {
  (void)in_sizes; (void)n_in; (void)out_size; (void)d_ws; (void)ws_size;
  const float* x = (const float*)d_in[0];
  float* out = (float*)d_out;
  const int grid = Bb * (Tt / TT);  // 2048 blocks
  log2quant_kernel<<<grid, NTHREADS, 0, stream>>>(x, out);
}